// QuantumHybridHead_65481071409940
// MI455X (gfx1250) — compile-verified
//
#include <hip/hip_runtime.h>

// CDNA5 (gfx1250) wave32 WMMA fragment types
typedef __attribute__((ext_vector_type(2))) float v2f;   // f32 A/B frag (16x4 / 4x16)
typedef __attribute__((ext_vector_type(4))) float v4f;
typedef __attribute__((ext_vector_type(8))) float v8f;   // f32 C/D frag (16x16)

#define IN_F           1024
#define ROWS_PER_WAVE  16
#define WAVES_PER_WG   8
#define ROWS_PER_WG    (ROWS_PER_WAVE * WAVES_PER_WG)   // 128
#define THREADS        (WAVES_PER_WG * 32)              // 256

// out[row] = cos( dot(x[row,:], W) + b ) * cos(phi)
__global__ __launch_bounds__(THREADS) void quantum_head_wmma(
    const float* __restrict__ x,    // [N, 1024]
    const float* __restrict__ W,    // [1024]
    const float* __restrict__ bias, // [1]
    const float* __restrict__ phi,  // [1]
    float* __restrict__ out)        // [N]
{
    __shared__ float sW[IN_F];                               // 4 KB, reused 256x
    __shared__ float sOut[WAVES_PER_WG][ROWS_PER_WAVE];

    const int tid = threadIdx.x;

    // Cooperative W preload: 256 threads x 4 floats = 1024 (global b128 -> ds b128)
    *(v4f*)(sW + tid * 4) = *(const v4f*)(W + tid * 4);
    __syncthreads();

    const int wave = tid >> 5;
    const int lane = tid & 31;
    const int m    = lane & 15;      // A-matrix row M (ISA layout: lanes 0-15 == 16-31)
    const int kh   = lane >> 4;      // K half: lanes 16-31 carry K+2,K+3

    const long row = (long)blockIdx.x * ROWS_PER_WG + (long)wave * ROWS_PER_WAVE + m;

    // Per-lane streams: both A (global) and B (LDS) fragments use index k0 + 2*kh
    const float* xp = x  + row * IN_F + 2 * kh;   // global, 8B-aligned float2 steps
    const float* wp = sW + 2 * kh;                // LDS

    // 4 accumulators break the WMMA->WMMA RAW dependence chain
    v8f c0 = {}, c1 = {}, c2 = {}, c3 = {};

#pragma unroll 4
    for (int k = 0; k < IN_F; k += 16) {
        v2f a0 = *(const v2f*)(xp + k);
        v2f a1 = *(const v2f*)(xp + k + 4);
        v2f a2 = *(const v2f*)(xp + k + 8);
        v2f a3 = *(const v2f*)(xp + k + 12);
        v2f b0 = *(const v2f*)(wp + k);
        v2f b1 = *(const v2f*)(wp + k + 4);
        v2f b2 = *(const v2f*)(wp + k + 8);
        v2f b3 = *(const v2f*)(wp + k + 12);
        // D = A(16x4 of x-rows) * B(4x16, W replicated in all columns) + C
        c0 = __builtin_amdgcn_wmma_f32_16x16x4_f32(false, a0, false, b0, (short)0, c0, false, false);
        c1 = __builtin_amdgcn_wmma_f32_16x16x4_f32(false, a1, false, b1, (short)0, c1, false, false);
        c2 = __builtin_amdgcn_wmma_f32_16x16x4_f32(false, a2, false, b2, (short)0, c2, false, false);
        c3 = __builtin_amdgcn_wmma_f32_16x16x4_f32(false, a3, false, b3, (short)0, c3, false, false);
    }

    v8f c = (c0 + c1) + (c2 + c3);   // elementwise VALU adds; layouts identical

    // C/D layout: lane l, VGPR v holds row (v + 8*(l>>4)), column (l&15).
    // All 16 columns are equal (B columns identical), so lanes 0 and 16 jointly
    // hold all 16 row results in their 8 C VGPRs.
    if (m == 0) {
        float* dst = &sOut[wave][kh * 8];
        dst[0] = c[0]; dst[1] = c[1]; dst[2] = c[2]; dst[3] = c[3];
        dst[4] = c[4]; dst[5] = c[5]; dst[6] = c[6]; dst[7] = c[7];
    }
    __syncthreads();

    if (lane < 16) {
        float theta = sOut[wave][lane] + bias[0];
        float r = __builtin_cosf(theta) * __builtin_cosf(phi[0]);
        out[(long)blockIdx.x * ROWS_PER_WG + (long)wave * ROWS_PER_WAVE + lane] = r;
    }
}

extern "C" void kernel_launch(void* const* d_in, const int* in_sizes, int n_in,
                              void* d_out, int out_size, void* d_ws, size_t ws_size,
                              hipStream_t stream) {
    const float* x    = (const float*)d_in[0];  // [N, 1024] f32
    const float* W    = (const float*)d_in[1];  // [1024]    f32
    const float* bias = (const float*)d_in[2];  // [1]       f32
    const float* phi  = (const float*)d_in[3];  // [1]       f32
    float* out = (float*)d_out;                 // [N]       f32

    const int N = out_size;                     // 262144; divisible by 128
    const int grid = N / ROWS_PER_WG;           // 2048 workgroups

    quantum_head_wmma<<<dim3(grid), dim3(THREADS), 0, stream>>>(x, W, bias, phi, out);
}